// GATClassifier_50826642981153
// MI455X (gfx1250) — compile-verified
//
#include <hip/hip_runtime.h>
#include <hip/hip_bf16.h>
#include <math.h>

// ---------------------------------------------------------------------------
// CDNA5 (gfx1250) GAT classifier.
//  - GEMMs use V_WMMA_F32_16X16X4_F32 (wave32, exact fp32 vs reference).
//  - Edge softmax/scatter via f32 global atomics (L2-resident: ~106MB scratch
//    fits in the 192MB global L2 of MI455X).
// ---------------------------------------------------------------------------

typedef __attribute__((ext_vector_type(2))) float v2f;
typedef __attribute__((ext_vector_type(8))) float v8f;

#define NEG_SLOPE 0.2f

// ---------------------------------------------------------------------------
// WMMA fp32 GEMM: C[M,N] = A[M,K] @ B[K,N]; one wave computes a 16x16 tile.
// A-frag (16x4): lanes 0-15 -> M=lane, K in {0,1}; lanes 16-31 -> K in {2,3}.
// B-frag (4x16): lanes 0-15 -> N=lane, K in {0,1}; lanes 16-31 -> K in {2,3}.
// C tile: vgpr v, lanes 0-15 -> M=v, N=lane; lanes 16-31 -> M=v+8.
// ---------------------------------------------------------------------------
__global__ void gemm16_wmma_f32(const float* __restrict__ A,
                                const float* __restrict__ B,
                                float* __restrict__ C,
                                int M, int K, int N) {
    const int wave          = threadIdx.x >> 5;
    const int lane          = threadIdx.x & 31;
    const int wavesPerBlock = blockDim.x >> 5;
    const int m0 = blockIdx.x << 4;
    const int n0 = (blockIdx.y * wavesPerBlock + wave) << 4;
    if (n0 >= N) return;

    const int half = lane >> 4;     // 0: K lanes {0,1}, 1: K lanes {2,3}
    const int lr   = lane & 15;
    int rowA = m0 + lr; if (rowA >= M) rowA = M - 1;   // clamp (tail tiles)
    const int colB = n0 + lr;

    v8f acc = {};
    for (int k0 = 0; k0 < K; k0 += 4) {
        const int kk = k0 + (half << 1);
        v2f a, b;
        a.x = A[(size_t)rowA * K + kk];
        a.y = A[(size_t)rowA * K + kk + 1];
        b.x = B[(size_t)kk       * N + colB];
        b.y = B[(size_t)(kk + 1) * N + colB];
        acc = __builtin_amdgcn_wmma_f32_16x16x4_f32(
                  false, a, false, b, (short)0, acc, false, false);
    }
#pragma unroll
    for (int v = 0; v < 8; ++v) {
        const int row = m0 + v + (half << 3);
        if (row < M) C[(size_t)row * N + n0 + lr] = acc[v];
    }
}

// ---------------------------------------------------------------------------
// Per-node attention scores: as[n,h] = dot(h[n, h*64:(h+1)*64], att_src[h,:])
// One wave per node; wave32 shfl_xor reduction within per-head lane groups.
// ---------------------------------------------------------------------------
__global__ void node_scores(const float* __restrict__ h,
                            const float* __restrict__ att_src,
                            const float* __restrict__ att_dst,
                            float* __restrict__ as_out,
                            float* __restrict__ ad_out,
                            int nNodes, int heads) {
    const int lane = threadIdx.x & 31;
    const int node = blockIdx.x * (blockDim.x >> 5) + (threadIdx.x >> 5);
    if (node >= nNodes) return;

    const int totCh     = heads * 64;
    const int chPerLane = totCh >> 5;        // heads*2
    const int groupSize = 32 / heads;        // lanes per head
    const int c0        = lane * chPerLane;  // flat index == head*64 + local

    float ss = 0.f, sd = 0.f;
    for (int i = 0; i < chPerLane; ++i) {
        const float v = h[(size_t)node * totCh + c0 + i];
        ss += v * att_src[c0 + i];
        sd += v * att_dst[c0 + i];
    }
    for (int m = groupSize >> 1; m >= 1; m >>= 1) {
        ss += __shfl_xor(ss, m, 32);
        sd += __shfl_xor(sd, m, 32);
    }
    if ((lane & (groupSize - 1)) == 0) {
        const int head = lane / groupSize;
        as_out[(size_t)node * heads + head] = ss;
        ad_out[(size_t)node * heads + head] = sd;
    }
}

// ---------------------------------------------------------------------------
// Float atomic max (works with -inf init; sign-aware int/uint ordering trick)
// ---------------------------------------------------------------------------
__device__ __forceinline__ void atomicMaxF(float* addr, float val) {
    if (val >= 0.0f) atomicMax((int*)addr, __float_as_int(val));
    else             atomicMin((unsigned int*)addr, __float_as_uint(val));
}

__device__ __forceinline__ void edge_sd(int e, const int* __restrict__ esrc,
                                        const int* __restrict__ edst, int E,
                                        int& s, int& d) {
    if (e < E) { s = esrc[e]; d = edst[e]; }
    else       { s = d = e - E; }            // self loops appended
}

__device__ __forceinline__ float edge_logit(const float* __restrict__ as,
                                            const float* __restrict__ ad,
                                            int s, int d, int hh, int heads) {
    const float v = as[(size_t)s * heads + hh] + ad[(size_t)d * heads + hh];
    return v > 0.f ? v : NEG_SLOPE * v;
}

__global__ void fillf(float* __restrict__ p, float v, int n) {
    const int i = blockIdx.x * blockDim.x + threadIdx.x;
    if (i < n) p[i] = v;
}

// Pass A: segment max of leaky-relu logits per (dst, head)
__global__ void edge_max(const float* __restrict__ as, const float* __restrict__ ad,
                         const int* __restrict__ esrc, const int* __restrict__ edst,
                         float* __restrict__ mmax, int E, int ET, int heads) {
    const int tid = blockIdx.x * blockDim.x + threadIdx.x;
    if (tid >= ET * heads) return;
    const int e  = tid / heads;
    const int hh = tid - e * heads;
    int s, d; edge_sd(e, esrc, edst, E, s, d);
    atomicMaxF(&mmax[(size_t)d * heads + hh], edge_logit(as, ad, s, d, hh, heads));
}

// Pass B: denom[dst,head] += exp(e - m[dst,head])
__global__ void edge_sum(const float* __restrict__ as, const float* __restrict__ ad,
                         const int* __restrict__ esrc, const int* __restrict__ edst,
                         const float* __restrict__ mmax, float* __restrict__ den,
                         int E, int ET, int heads) {
    const int tid = blockIdx.x * blockDim.x + threadIdx.x;
    if (tid >= ET * heads) return;
    const int e  = tid / heads;
    const int hh = tid - e * heads;
    int s, d; edge_sd(e, esrc, edst, E, s, d);
    const float lr = edge_logit(as, ad, s, d, hh, heads);
    const float p  = expf(lr - mmax[(size_t)d * heads + hh]);
    atomicAdd(&den[(size_t)d * heads + hh], p);
}

// Pass C: agg[dst,:] += alpha * h[src,:]  (float4 gather, f32 atomic scatter)
__global__ void edge_scatter(const float* __restrict__ h,
                             const float* __restrict__ as, const float* __restrict__ ad,
                             const int* __restrict__ esrc, const int* __restrict__ edst,
                             const float* __restrict__ mmax, const float* __restrict__ den,
                             float* __restrict__ agg, int E, int ET, int heads) {
    const int chunks = heads * 16;           // (heads*64)/4 float4 chunks per edge
    const long tid = (long)blockIdx.x * blockDim.x + threadIdx.x;
    if (tid >= (long)ET * chunks) return;
    const int e  = (int)(tid / chunks);
    const int q  = (int)(tid - (long)e * chunks);
    const int c0 = q << 2;
    const int hh = c0 >> 6;
    int s, d; edge_sd(e, esrc, edst, E, s, d);

    const float lr    = edge_logit(as, ad, s, d, hh, heads);
    const float p     = expf(lr - mmax[(size_t)d * heads + hh]);
    const float alpha = p / den[(size_t)d * heads + hh];

    const int totCh = heads * 64;
    const float4 hv = *reinterpret_cast<const float4*>(h + (size_t)s * totCh + c0);
    float* dp = agg + (size_t)d * totCh + c0;
    atomicAdd(dp + 0, hv.x * alpha);
    atomicAdd(dp + 1, hv.y * alpha);
    atomicAdd(dp + 2, hv.z * alpha);
    atomicAdd(dp + 3, hv.w * alpha);
}

// out = elu(agg + bias), in place
__global__ void bias_elu(float* __restrict__ agg, const float* __restrict__ bias,
                         int nNodes, int ch) {
    const long i = (long)blockIdx.x * blockDim.x + threadIdx.x;
    if (i >= (long)nNodes * ch) return;
    const int c = (int)(i % ch);
    const float v = agg[i] + bias[c];
    agg[i] = v > 0.f ? v : (expf(v) - 1.f);
}

// global mean pool: sums[g,c] += h[n,c]; cnt[g] += 1
__global__ void pool_sum(const float* __restrict__ h, const int* __restrict__ batch,
                         float* __restrict__ sums, float* __restrict__ cnt, int nNodes) {
    const long tid = (long)blockIdx.x * blockDim.x + threadIdx.x;
    if (tid >= (long)nNodes * 64) return;
    const int n = (int)(tid >> 6);
    const int c = (int)(tid & 63);
    const int g = batch[n];
    atomicAdd(&sums[(size_t)g * 64 + c], h[(size_t)n * 64 + c]);
    if (c == 0) atomicAdd(&cnt[g], 1.0f);
}

// out[g,k] = (sums[g,:]/max(cnt,1)) @ fc_w[:,k] + fc_b[k]; 32 graphs x 2
__global__ void head_linear(const float* __restrict__ sums, const float* __restrict__ cnt,
                            const float* __restrict__ fcw, const float* __restrict__ fcb,
                            float* __restrict__ out) {
    const int t = threadIdx.x;
    if (t >= 64) return;
    const int g = t >> 1, k = t & 1;
    const float inv = 1.0f / fmaxf(cnt[g], 1.0f);
    float acc = fcb[k];
    for (int c = 0; c < 64; ++c)
        acc += (sums[g * 64 + c] * inv) * fcw[c * 2 + k];
    out[g * 2 + k] = acc;
}

// ---------------------------------------------------------------------------
extern "C" void kernel_launch(void* const* d_in, const int* in_sizes, int n_in,
                              void* d_out, int out_size, void* d_ws, size_t ws_size,
                              hipStream_t stream) {
    const float* x    = (const float*)d_in[0];
    const int*   esrc = (const int*)  d_in[1];
    const int*   edst = (const int*)  d_in[2];
    // d_in[3] edge_weight: ignored (edge_dim=None in reference)
    const int*   batch= (const int*)  d_in[4];
    const float* W1   = (const float*)d_in[5];
    const float* aS1  = (const float*)d_in[6];
    const float* aD1  = (const float*)d_in[7];
    const float* b1   = (const float*)d_in[8];
    const float* W2   = (const float*)d_in[9];
    const float* aS2  = (const float*)d_in[10];
    const float* aD2  = (const float*)d_in[11];
    const float* b2   = (const float*)d_in[12];
    const float* fcw  = (const float*)d_in[13];
    const float* fcb  = (const float*)d_in[14];
    float* out = (float*)d_out;

    const int NN = in_sizes[0] / 256;   // 50000
    const int NE = in_sizes[1];         // 800000
    const int ET = NE + NN;             // + self loops

    // ---- scratch layout (float offsets; 256B aligned). Peak ~106MB < L2. ----
    auto al = [](size_t n) { return (n + 63) & ~(size_t)63; };
    const size_t H1 = (size_t)NN * 256;
    size_t o = 0;
    const size_t off_h1   = o; o += H1;                 // GEMM1 out [N,256]
    const size_t off_agg1 = o; o += H1;                 // agg1 -> elu -> layer2 in
    const size_t off_as1  = o; o += al((size_t)NN * 4);
    const size_t off_ad1  = o; o += al((size_t)NN * 4);
    const size_t off_m1   = o; o += al((size_t)NN * 4);
    const size_t off_den1 = o; o += al((size_t)NN * 4);
    // phase 2 reuses the retired h1 region [0, H1)
    size_t p = 0;
    const size_t off_h2   = p; p += (size_t)NN * 64;    // GEMM2 out [N,64]
    const size_t off_as2  = p; p += al((size_t)NN);
    const size_t off_ad2  = p; p += al((size_t)NN);
    const size_t off_m2   = p; p += al((size_t)NN);
    const size_t off_den2 = p; p += al((size_t)NN);
    const size_t off_agg2 = p; p += (size_t)NN * 64;    // agg2 -> elu -> pooled in
    const size_t off_sums = p;
    const size_t off_cnt  = off_sums + 2048;            // 32*64 sums, then 32 cnts

    float* ws = (float*)d_ws;
    auto cdiv = [](long a, long b) { return (int)((a + b - 1) / b); };

    // ================= Layer 1 (heads=4, ch=64) =================
    {
        dim3 grid(cdiv(NN, 16), 2);                     // 16 col tiles / 8 waves
        gemm16_wmma_f32<<<grid, 256, 0, stream>>>(x, W1, ws + off_h1, NN, 256, 256);
    }
    node_scores<<<cdiv(NN, 8), 256, 0, stream>>>(ws + off_h1, aS1, aD1,
                                                 ws + off_as1, ws + off_ad1, NN, 4);
    fillf<<<cdiv((long)NN * 4, 256), 256, 0, stream>>>(ws + off_m1, -INFINITY, NN * 4);
    edge_max<<<cdiv((long)ET * 4, 256), 256, 0, stream>>>(
        ws + off_as1, ws + off_ad1, esrc, edst, ws + off_m1, NE, ET, 4);
    hipMemsetAsync(ws + off_den1, 0, (size_t)NN * 4 * sizeof(float), stream);
    edge_sum<<<cdiv((long)ET * 4, 256), 256, 0, stream>>>(
        ws + off_as1, ws + off_ad1, esrc, edst, ws + off_m1, ws + off_den1, NE, ET, 4);
    hipMemsetAsync(ws + off_agg1, 0, H1 * sizeof(float), stream);
    edge_scatter<<<cdiv((long)ET * 64, 256), 256, 0, stream>>>(
        ws + off_h1, ws + off_as1, ws + off_ad1, esrc, edst,
        ws + off_m1, ws + off_den1, ws + off_agg1, NE, ET, 4);
    bias_elu<<<cdiv(H1, 256), 256, 0, stream>>>(ws + off_agg1, b1, NN, 256);

    // ================= Layer 2 (heads=1, ch=64) =================
    {
        dim3 grid(cdiv(NN, 16), 1);                     // 4 col tiles / 4 waves
        gemm16_wmma_f32<<<grid, 128, 0, stream>>>(ws + off_agg1, W2, ws + off_h2,
                                                  NN, 256, 64);
    }
    node_scores<<<cdiv(NN, 8), 256, 0, stream>>>(ws + off_h2, aS2, aD2,
                                                 ws + off_as2, ws + off_ad2, NN, 1);
    fillf<<<cdiv(NN, 256), 256, 0, stream>>>(ws + off_m2, -INFINITY, NN);
    edge_max<<<cdiv(ET, 256), 256, 0, stream>>>(
        ws + off_as2, ws + off_ad2, esrc, edst, ws + off_m2, NE, ET, 1);
    hipMemsetAsync(ws + off_den2, 0, (size_t)NN * sizeof(float), stream);
    edge_sum<<<cdiv(ET, 256), 256, 0, stream>>>(
        ws + off_as2, ws + off_ad2, esrc, edst, ws + off_m2, ws + off_den2, NE, ET, 1);
    hipMemsetAsync(ws + off_agg2, 0, (size_t)NN * 64 * sizeof(float), stream);
    edge_scatter<<<cdiv((long)ET * 16, 256), 256, 0, stream>>>(
        ws + off_h2, ws + off_as2, ws + off_ad2, esrc, edst,
        ws + off_m2, ws + off_den2, ws + off_agg2, NE, ET, 1);
    bias_elu<<<cdiv((long)NN * 64, 256), 256, 0, stream>>>(ws + off_agg2, b2, NN, 64);

    // ================= Pool + head =================
    hipMemsetAsync(ws + off_sums, 0, (2048 + 64) * sizeof(float), stream);
    pool_sum<<<cdiv((long)NN * 64, 256), 256, 0, stream>>>(
        ws + off_agg2, batch, ws + off_sums, ws + off_cnt, NN);
    head_linear<<<1, 64, 0, stream>>>(ws + off_sums, ws + off_cnt, fcw, fcb, out);
}